// ScopeEnhancedDualEncoder_82231443849913
// MI455X (gfx1250) — compile-verified
//
#include <hip/hip_runtime.h>
#include <hip/hip_bf16.h>
#include <math.h>

typedef __attribute__((ext_vector_type(2))) float v2f;
typedef __attribute__((ext_vector_type(8))) float v8f;

#define DH    768
#define SMAX  12
#define WMAXW 400
#define CHUNK 32

// ---------------------------------------------------------------------------
// Kernel 1: segment sums into [M, W, D] + counts [M, W], LDS-accumulated.
// grid = (D/CHUNK, M), block = 256 (32 token-lanes x 8 quads of 4 dims).
// 128-bit coalesced global loads; ds_add_f32 accumulation; no global atomics.
// ---------------------------------------------------------------------------
__global__ __launch_bounds__(256)
void seg_sum_kernel(const float* __restrict__ hidden,
                    const int*   __restrict__ word_ids,
                    float*       __restrict__ sums,
                    float*       __restrict__ counts,
                    int L, int W)
{
    __shared__ float s_acc[WMAXW * CHUNK];   // 50 KB
    __shared__ float s_cnt[WMAXW];

    const int tid     = threadIdx.x;
    const int m       = blockIdx.y;
    const int chunk   = blockIdx.x;
    const int dQuad   = (tid & 7) << 2;      // 0,4,...,28
    const int tokLane = tid >> 3;            // 0..31

    for (int i = tid; i < W * (CHUNK / 4); i += 256)
        *(float4*)&s_acc[i * 4] = make_float4(0.f, 0.f, 0.f, 0.f);
    for (int i = tid; i < W; i += 256) s_cnt[i] = 0.f;
    __syncthreads();

    const float* hbase = hidden + (size_t)m * L * DH + chunk * CHUNK;
    const int*   wbase = word_ids + (size_t)m * L;

    for (int tok = tokLane; tok < L; tok += 32) {
        int    wid = wbase[tok];
        float4 v   = *(const float4*)(hbase + (size_t)tok * DH + dQuad);
        float* dst = &s_acc[wid * CHUNK + dQuad];
        atomicAdd(dst + 0, v.x);
        atomicAdd(dst + 1, v.y);
        atomicAdd(dst + 2, v.z);
        atomicAdd(dst + 3, v.w);
        if (chunk == 0 && (tid & 7) == 0) atomicAdd(&s_cnt[wid], 1.f);
    }
    __syncthreads();

    float* obase = sums + (size_t)m * W * DH + chunk * CHUNK;
    for (int i = tid; i < W * (CHUNK / 4); i += 256) {
        int w = i >> 3, q = (i & 7) << 2;
        *(float4*)(obase + (size_t)w * DH + q) = *(float4*)&s_acc[w * CHUNK + q];
    }
    if (chunk == 0)
        for (int w = tid; w < W; w += 256)
            counts[(size_t)m * W + w] = s_cnt[w];
}

// ---------------------------------------------------------------------------
// Kernel 2: per-sample scope projection (WMMA f32 16x16x4), logits, importance,
// masked softmax over W, and pooled output. grid = M, block = 256 (8 waves).
// Inner K-loop: branch-free b64 loads (global A, LDS B-transpose), EXEC all-1s.
// ---------------------------------------------------------------------------
__global__ __launch_bounds__(256)
void scope_pool_kernel(const float* __restrict__ sums,
                       const float* __restrict__ counts,
                       const float* __restrict__ proj_w,     // [768,12] row-major
                       const float* __restrict__ proj_b,     // [12]
                       const float* __restrict__ scope_imp,  // [12]
                       float*       __restrict__ logits_out, // [M,W,12] or null
                       float*       __restrict__ pooled_out, // [M,768]
                       int W)
{
    __shared__ __align__(16) float s_pwT[16 * DH];   // 48 KB, B transposed+padded
    __shared__ float s_pb[16];
    __shared__ float s_si[16];
    __shared__ float s_rcp[WMAXW];
    __shared__ float s_maskf[WMAXW];
    __shared__ float s_impw[WMAXW];
    __shared__ float s_red[256];
    __shared__ float s_tile[8][16][16];              // per-wave logits tile
    __shared__ float s_invsum;

    const int tid  = threadIdx.x;
    const int m    = blockIdx.x;
    const int lane = tid & 31;
    const int wv   = tid >> 5;

    // B transpose: s_pwT[n*768 + k] = (n < 12) ? proj_w[k*12 + n] : 0
    for (int i = tid; i < 16 * DH; i += 256) {
        int n = i / DH, k = i - n * DH;
        s_pwT[i] = (n < SMAX) ? proj_w[k * SMAX + n] : 0.f;
    }
    if (tid < 16) {
        s_pb[tid] = (tid < SMAX) ? proj_b[tid]    : 0.f;
        s_si[tid] = (tid < SMAX) ? scope_imp[tid] : 0.f;
    }
    for (int w = tid; w < W; w += 256) {
        float c    = counts[(size_t)m * W + w];
        s_maskf[w] = (c > 0.f) ? 1.f : 0.f;
        s_rcp[w]   = 1.f / fmaxf(c, 1.f);
    }
    __syncthreads();

    const int    nTiles = (W + 15) >> 4;
    const float* base   = sums + (size_t)m * W * DH;

    for (int t = wv; t < nTiles; t += 8) {
        const int wBase = t << 4;
        // A: lanes 0-15 -> rows M with K0/K1, lanes 16-31 -> K2/K3 (clamped row)
        const int wA   = wBase + (lane & 15);
        const int wClp = (wA < W) ? wA : 0;
        const int kOff = (lane >> 4) << 1;           // 0 or 2
        const float* rowA = base + (size_t)wClp * DH + kOff;
        // B: lanes 0-15 -> cols N with K0/K1, lanes 16-31 -> K2/K3
        const float* colB = &s_pwT[(lane & 15) * DH + kOff];

        v8f acc = {};
        #pragma unroll 8
        for (int k0 = 0; k0 < DH; k0 += 4) {
            v2f a = *(const v2f*)(rowA + k0);        // global_load_b64
            v2f b = *(const v2f*)(colB + k0);        // ds_load_b64
            acc = __builtin_amdgcn_wmma_f32_16x16x4_f32(
                false, a, false, b, (short)0, acc, false, false);
        }

        // C/D layout: VGPR r = (M=r, N=lane) lanes 0-15; (M=r+8, N=lane-16) else.
        const int rowOff = (lane >> 4) << 3;         // 0 or 8
        const int col    = lane & 15;
        #pragma unroll
        for (int r = 0; r < 8; ++r) {
            int   row = r + rowOff;
            int   w   = wBase + row;
            float v   = 0.f;
            if (w < W)
                v = (acc[r] * s_rcp[w] + s_pb[col]) * s_maskf[w];
            s_tile[wv][row][col] = v;
            if (logits_out && w < W && col < SMAX)
                logits_out[(size_t)(m * W + w) * SMAX + col] = v;
        }

        // per-word S-softmax -> importance (lanes 0..15, row = lane)
        if (lane < 16) {
            int w = wBase + lane;
            if (w < W) {
                if (s_maskf[w] > 0.f) {
                    float mx = -__builtin_inff();
                    for (int s = 0; s < SMAX; ++s)
                        mx = fmaxf(mx, s_tile[wv][lane][s]);
                    float z = 0.f, dot = 0.f;
                    for (int s = 0; s < SMAX; ++s) {
                        float e = expf(s_tile[wv][lane][s] - mx);
                        z += e; dot += e * s_si[s];
                    }
                    s_impw[w] = dot / z;
                } else {
                    s_impw[w] = -1e4f;   // NEG_BIG
                }
            }
        }
    }
    __syncthreads();

    // masked softmax over W (block reduction)
    float lmax = -__builtin_inff();
    for (int w = tid; w < W; w += 256) lmax = fmaxf(lmax, s_impw[w]);
    s_red[tid] = lmax; __syncthreads();
    for (int off = 128; off > 0; off >>= 1) {
        if (tid < off) s_red[tid] = fmaxf(s_red[tid], s_red[tid + off]);
        __syncthreads();
    }
    float gmax = s_red[0]; __syncthreads();

    float lsum = 0.f;
    for (int w = tid; w < W; w += 256) {
        float e = expf(s_impw[w] - gmax);
        s_impw[w] = e; lsum += e;
    }
    s_red[tid] = lsum; __syncthreads();
    for (int off = 128; off > 0; off >>= 1) {
        if (tid < off) s_red[tid] += s_red[tid + off];
        __syncthreads();
    }
    if (tid == 0) s_invsum = 1.f / s_red[0];
    __syncthreads();

    // fold weight * rcp * invZ into s_impw once
    const float inv = s_invsum;
    for (int w = tid; w < W; w += 256)
        s_impw[w] = s_impw[w] * inv * s_rcp[w];
    __syncthreads();

    // pooled[d] = sum_w s_impw[w] * sums[w,d]   (coalesced over d)
    for (int d = tid; d < DH; d += 256) {
        float a2 = 0.f;
        for (int w = 0; w < W; ++w)
            a2 += s_impw[w] * base[(size_t)w * DH + d];
        pooled_out[(size_t)m * DH + d] = a2;
    }
}

// ---------------------------------------------------------------------------
extern "C" void kernel_launch(void* const* d_in, const int* in_sizes, int n_in,
                              void* d_out, int out_size, void* d_ws, size_t ws_size,
                              hipStream_t stream) {
    (void)in_sizes; (void)n_in; (void)out_size; (void)ws_size;

    const float* q_hidden   = (const float*)d_in[0];   // [16,256,768]
    const float* pos_hidden = (const float*)d_in[1];   // [16,512,768]
    const float* neg_hidden = (const float*)d_in[2];   // [16,7,512,768]
    const float* proj_w     = (const float*)d_in[3];   // [768,12]
    const float* proj_b     = (const float*)d_in[4];   // [12]
    const float* scope_imp  = (const float*)d_in[5];   // [12]
    const int*   q_wid      = (const int*)d_in[6];     // [16,256]
    const int*   p_wid      = (const int*)d_in[7];     // [16,512]
    const int*   n_wid      = (const int*)d_in[8];     // [16,7,512]

    const int B = 16, N = 7, Lq = 256, Lp = 512;
    const int Wq = 200, Wp = 400;
    const int Mq = B, Mp = B, Mn = B * N;

    // workspace layout (floats)
    float* ws     = (float*)d_ws;
    float* sums_q = ws;                                 // 16*200*768
    float* cnt_q  = sums_q + (size_t)Mq * Wq * DH;      // 16*200
    float* sums_p = cnt_q  + (size_t)Mq * Wq;           // 16*400*768
    float* cnt_p  = sums_p + (size_t)Mp * Wp * DH;      // 16*400
    float* sums_n = cnt_p  + (size_t)Mp * Wp;           // 112*400*768
    float* cnt_n  = sums_n + (size_t)Mn * Wp * DH;      // 112*400

    // output layout (floats): q_pooled, p_pooled, q_logits, p_logits, n_pooled
    float* out      = (float*)d_out;
    float* q_pooled = out;                              // 16*768
    float* p_pooled = q_pooled + (size_t)Mq * DH;       // 16*768
    float* q_logits = p_pooled + (size_t)Mp * DH;       // 16*200*12
    float* p_logits = q_logits + (size_t)Mq * Wq * SMAX;// 16*400*12
    float* n_pooled = p_logits + (size_t)Mp * Wp * SMAX;// 16*7*768

    const int nChunks = DH / CHUNK;  // 24

    seg_sum_kernel<<<dim3(nChunks, Mq), 256, 0, stream>>>(
        q_hidden, q_wid, sums_q, cnt_q, Lq, Wq);
    seg_sum_kernel<<<dim3(nChunks, Mp), 256, 0, stream>>>(
        pos_hidden, p_wid, sums_p, cnt_p, Lp, Wp);
    seg_sum_kernel<<<dim3(nChunks, Mn), 256, 0, stream>>>(
        neg_hidden, n_wid, sums_n, cnt_n, Lp, Wp);

    scope_pool_kernel<<<Mq, 256, 0, stream>>>(
        sums_q, cnt_q, proj_w, proj_b, scope_imp, q_logits, q_pooled, Wq);
    scope_pool_kernel<<<Mp, 256, 0, stream>>>(
        sums_p, cnt_p, proj_w, proj_b, scope_imp, p_logits, p_pooled, Wp);
    scope_pool_kernel<<<Mn, 256, 0, stream>>>(
        sums_n, cnt_n, proj_w, proj_b, scope_imp, nullptr, n_pooled, Wp);
}